// PFLASoftmax_38225208934523
// MI455X (gfx1250) — compile-verified
//
#include <hip/hip_runtime.h>
#include <hip/hip_bf16.h>

// PFLA + row-normalize, fp32, rows of 1024.
// HBM-bound (805 MB compulsory traffic @ 23.3 TB/s ~= 35 us). Pipeline:
//   global_load_async_to_lds_b128  (row r+1, ASYNCcnt)   overlaps
//   ds_load_b128 -> VALU -> wave32 reduce -> NT b128 store (row r)
// 4 rows per block amortize knot-table setup; each lane stages/consumes only
// its own 16B LDS region, so the async pipeline needs no extra barriers.

typedef float v4f __attribute__((ext_vector_type(4)));

#define ROW_LEN   1024
#define BLOCK     256          // 8 wave32; 256 lanes * 16B = one 4KB row
#define ROWS_PB   4
#define MAX_KNOTS 64

__device__ __forceinline__ void stage_row_async(unsigned lds_addr, unsigned long long gbase,
                                                unsigned goff) {
    // GVS form: mem = SADDR64 + VADDR32 (+0). VDST = per-lane LDS byte address.
    asm volatile("global_load_async_to_lds_b128 %0, %1, %2"
                 :: "v"(lds_addr), "v"(goff), "s"(gbase)
                 : "memory");
}

__device__ __forceinline__ void wait_async0() {
#if __has_builtin(__builtin_amdgcn_s_wait_asynccnt)
    __builtin_amdgcn_s_wait_asynccnt(0);
#else
    asm volatile("s_wait_asynccnt 0x0" ::: "memory");
#endif
}

__global__ __launch_bounds__(BLOCK) void pfla_norm_kernel(
    const float* __restrict__ x,
    const float* __restrict__ x_vals,
    const float* __restrict__ y_vals,
    float* __restrict__ out,
    int n_knots, int n_rows)
{
    __shared__ float s_xk[MAX_KNOTS];
    __shared__ float s_yk[MAX_KNOTS];
    __shared__ float s_sl[MAX_KNOTS];
    __shared__ __align__(16) float s_buf[2][ROW_LEN];   // async-staged row double buffer
    __shared__ float s_part[BLOCK / 32];
    __shared__ float s_inv;

    const int t  = threadIdx.x;
    const int nk = n_knots;

    // ---- knot tables (once per block, amortized over ROWS_PB rows) ----
    if (t < nk) {
        const float xv = x_vals[t];
        const float yv = y_vals[t];
        s_xk[t] = xv;
        s_yk[t] = yv * yv;                      // square knots once
    }
    __syncthreads();
    if (t < nk) {
        const int j = min(t + 1, nk - 1);
        s_sl[t] = (s_yk[j] - s_yk[t]) / fmaxf(s_xk[j] - s_xk[t], 1e-6f);
    }
    __syncthreads();

    const float xk0   = s_xk[0];
    const float inv_h = (float)(nk - 1) / (s_xk[nk - 1] - xk0);

    const long long row0      = (long long)blockIdx.x * ROWS_PB;
    const int       rows_here = (int)min((long long)ROWS_PB, (long long)n_rows - row0);
    if (rows_here <= 0) return;

    const unsigned long long xbase  = (unsigned long long)x;
    const unsigned           lane_b = (unsigned)t * 16u;                  // byte offset in row
    const unsigned lds_buf0 = (unsigned)(uintptr_t)&s_buf[0][0] + lane_b; // raw LDS byte addr
    const unsigned lds_buf1 = (unsigned)(uintptr_t)&s_buf[1][0] + lane_b;

    // prime the pipeline: stage row 0
    stage_row_async(lds_buf0, xbase, (unsigned)(row0 * ROW_LEN * 4) + lane_b);

    for (int r = 0; r < rows_here; ++r) {
        const int bi = r & 1;

        wait_async0();                           // row r is in LDS
        v4f v = *(const v4f*)&s_buf[bi][t * 4];  // ds_load_b128 (own 16B region)

        // ---- piecewise-linear interp + local sum (forces DScnt wait on v) ----
        v4f   y;
        float local = 0.0f;
#pragma unroll
        for (int c = 0; c < 4; ++c) {
            const float xi  = v[c];
            const bool  fin = __builtin_isfinite(xi);
            const float xs  = fin ? xi : 0.0f;

            int i = (int)ceilf((xs - xk0) * inv_h);          // searchsorted, uniform knots
            i = min(max(i, 0), nk - 1);
            if (i < nk - 1 && s_xk[i] < xs)      ++i;        // +-1 fixup vs exact fp32 knots
            else if (i > 0 && s_xk[i - 1] >= xs) --i;

            float yv = fmaf(s_sl[i], xs - s_xk[i], s_yk[i]);
            yv   = fin ? yv : 0.0f;
            y[c] = yv;
            local += yv;
        }

        // row r fully consumed by this lane -> safe to overwrite other buffer:
        // prefetch row r+1 while we reduce/store row r.
        if (r + 1 < rows_here) {
            stage_row_async(bi ? lds_buf0 : lds_buf1, xbase,
                            (unsigned)((row0 + r + 1) * ROW_LEN * 4) + lane_b);
        }

        // ---- wave32 reduce + block combine ----
#pragma unroll
        for (int off = 16; off > 0; off >>= 1)
            local += __shfl_down(local, off, 32);
        if ((t & 31) == 0) s_part[t >> 5] = local;
        __syncthreads();
        if (t == 0) {
            float tot = 0.0f;
#pragma unroll
            for (int w = 0; w < BLOCK / 32; ++w) tot += s_part[w];
            s_inv = 1.0f / (tot + 1e-6f);
        }
        __syncthreads();
        const float inv = s_inv;

        v4f o;
        o[0] = y[0] * inv;
        o[1] = y[1] * inv;
        o[2] = y[2] * inv;
        o[3] = y[3] * inv;
        v4f* yout = (v4f*)(out + (size_t)(row0 + r) * ROW_LEN);
        __builtin_nontemporal_store(o, yout + t);            // NT: don't rinse 402MB thru L2
    }
}

extern "C" void kernel_launch(void* const* d_in, const int* in_sizes, int n_in,
                              void* d_out, int out_size, void* d_ws, size_t ws_size,
                              hipStream_t stream) {
    (void)n_in; (void)out_size; (void)d_ws; (void)ws_size;

    const float* x      = (const float*)d_in[0];
    const float* x_vals = (const float*)d_in[1];
    const float* y_vals = (const float*)d_in[2];
    float*       out    = (float*)d_out;

    const int total   = in_sizes[0];             // 8*12*1024*1024
    const int n_knots = in_sizes[1];             // 30
    const int n_rows  = total / ROW_LEN;         // 98304
    const int blocks  = (n_rows + ROWS_PB - 1) / ROWS_PB;

    pfla_norm_kernel<<<blocks, BLOCK, 0, stream>>>(x, x_vals, y_vals, out, n_knots, n_rows);
}